// GAT_68118181314632
// MI455X (gfx1250) — compile-verified
//
#include <hip/hip_runtime.h>
#include <hip/hip_bf16.h>

typedef __attribute__((ext_vector_type(2))) float v2f;
typedef __attribute__((ext_vector_type(8))) float v8f;

#define NEG_SLOPE 0.2f

// ---------------------------------------------------------------------------
// Dense H = X @ W via V_WMMA_F32_16X16X4_F32 (exact fp32 matrix path).
// One wave computes one 16x16 output tile; 8 waves / block.
// A (16x4 f32): lanes 0-15 hold M=lane {K=k, k+1}; lanes 16-31 hold {k+2, k+3}.
// B (4x16 f32): v0 = {K=k (lanes 0-15), K=k+2 (lanes 16-31)}, v1 = {k+1, k+3}.
// C/D (16x16 f32): VGPR r -> M=r (lanes 0-15) / M=8+r (lanes 16-31), N=lane&15.
// ---------------------------------------------------------------------------
template <int CIN, int COUT>
__global__ __launch_bounds__(256) void gemm_wmma(const float* __restrict__ X,
                                                 const float* __restrict__ W,
                                                 float* __restrict__ H, int n) {
  constexpr int COLT = COUT / 16;  // column tiles
  constexpr int ROWT = 8 / COLT;   // row tiles per block (8 waves)
  const int wave = threadIdx.x >> 5;
  const int lane = threadIdx.x & 31;
  const int rowTile = blockIdx.x * ROWT + wave / COLT;
  const int colTile = wave % COLT;
  const int nRowTiles = (n + 15) >> 4;
  if (rowTile >= nRowTiles) return;  // wave-uniform: EXEC stays all-1s

  const int m = lane & 15;
  const int hi = lane >> 4;  // K half select
  int row = rowTile * 16 + m;
  if (row >= n) row = n - 1;  // clamp loads for a ragged tail (stores guarded)
  const float* __restrict__ xr = X + (size_t)row * CIN;
  const float* __restrict__ wc = W + colTile * 16 + m;

  v8f acc = {};
#pragma unroll 4
  for (int k = 0; k < CIN; k += 4) {
    const int ka = k + hi * 2;
    v2f A, B;
    const float2 a2 = *(const float2*)(xr + ka);  // global_load_b64
    A[0] = a2.x;
    A[1] = a2.y;
    B[0] = wc[(size_t)ka * COUT];
    B[1] = wc[(size_t)(ka + 1) * COUT];
    acc = __builtin_amdgcn_wmma_f32_16x16x4_f32(false, A, false, B, (short)0,
                                                acc, false, false);
  }

  const int rbase = rowTile * 16 + hi * 8;
  float* __restrict__ hr = H + (size_t)rbase * COUT + colTile * 16 + m;
  if (rowTile * 16 + 16 <= n) {
    // Full tile (always the case when n % 16 == 0): unconditional stores,
    // wave-uniform branch -> no per-lane exec juggling.
#pragma unroll
    for (int r = 0; r < 8; ++r) hr[(size_t)r * COUT] = acc[r];
  } else {
#pragma unroll
    for (int r = 0; r < 8; ++r)
      if (rbase + r < n) hr[(size_t)r * COUT] = acc[r];
  }
}

// ---------------------------------------------------------------------------
// Per-node attention logits: alpha_src[i] = h[i,:].a_src ; alpha_dst likewise.
// One wave32 per node, shuffle-tree reduction.
// ---------------------------------------------------------------------------
template <int C>
__global__ __launch_bounds__(256) void alpha_kernel(
    const float* __restrict__ H, const float* __restrict__ a_src,
    const float* __restrict__ a_dst, float* __restrict__ asrc,
    float* __restrict__ adst, int n) {
  const int node = (int)((blockIdx.x * (size_t)blockDim.x + threadIdx.x) >> 5);
  const int lane = threadIdx.x & 31;
  if (node >= n) return;
  const float* __restrict__ h = H + (size_t)node * C;
  float s = 0.f, d = 0.f;
#pragma unroll
  for (int c = lane; c < C; c += 32) {
    const float hv = h[c];
    s += hv * a_src[c];
    d += hv * a_dst[c];
  }
#pragma unroll
  for (int off = 16; off > 0; off >>= 1) {
    s += __shfl_xor(s, off, 32);
    d += __shfl_xor(d, off, 32);
  }
  if (lane == 0) {
    asrc[node] = s;
    adst[node] = d;
  }
}

// ---------------------------------------------------------------------------
// Edge score + segment-max. Float max via ordered int-punning atomics:
// memory pre-set to 0xFFFFFFFF (below everything in this ordering).
// ---------------------------------------------------------------------------
__global__ __launch_bounds__(256) void edge_score_max(
    const int* __restrict__ src, const int* __restrict__ dst,
    const float* __restrict__ asrc, const float* __restrict__ adst,
    float* __restrict__ e, int* __restrict__ m, int E) {
  const int i = blockIdx.x * blockDim.x + threadIdx.x;
  if (i >= E) return;
  const int d = dst[i];
  float v = asrc[src[i]] + adst[d];
  v = (v > 0.f) ? v : NEG_SLOPE * v;
  e[i] = v;
  if (v >= 0.f)
    atomicMax(&m[d], __float_as_int(v));
  else
    atomicMin((unsigned int*)&m[d], __float_as_uint(v));
}

// ---------------------------------------------------------------------------
// p = exp(e - m[dst]) (stored in-place) ; denom[dst] += p
// ---------------------------------------------------------------------------
__global__ __launch_bounds__(256) void edge_exp_sum(const int* __restrict__ dst,
                                                    float* __restrict__ e,
                                                    const float* __restrict__ m,
                                                    float* __restrict__ den,
                                                    int E) {
  const int i = blockIdx.x * blockDim.x + threadIdx.x;
  if (i >= E) return;
  const int d = dst[i];
  const float p = __expf(e[i] - m[d]);
  e[i] = p;
  atomicAdd(&den[d], p);
}

// ---------------------------------------------------------------------------
// acc[dst,:] += h[src,:] * (p / (denom[dst] + 1e-16)). One wave per edge;
// lanes cover channel pairs (float2 gather = global_load_b64, atomic scatter
// stays L2-resident: acc arrays are 12.8 MB << 192 MB L2).
// ---------------------------------------------------------------------------
template <int C>
__global__ __launch_bounds__(256) void aggregate(
    const int* __restrict__ src, const int* __restrict__ dst,
    const float* __restrict__ e, const float* __restrict__ den,
    const float* __restrict__ H, float* __restrict__ acc, int E) {
  const long long gw = (blockIdx.x * (size_t)blockDim.x + threadIdx.x) >> 5;
  const int lane = threadIdx.x & 31;
  const long long nw = ((long long)gridDim.x * blockDim.x) >> 5;
  for (long long i = gw; i < E; i += nw) {
    const int s = src[i];
    const int d = dst[i];
    const float a = e[i] / (den[d] + 1e-16f);
    const float* __restrict__ hs = H + (size_t)s * C;
    float* __restrict__ od = acc + (size_t)d * C;
#pragma unroll
    for (int c = lane * 2; c < C; c += 64) {
      const float2 hv = *(const float2*)(hs + c);  // global_load_b64
      atomicAdd(&od[c], hv.x * a);
      atomicAdd(&od[c + 1], hv.y * a);
    }
  }
}

// ---------------------------------------------------------------------------
// out = (relu?)(acc + bias)
// ---------------------------------------------------------------------------
template <int C, bool RELU>
__global__ __launch_bounds__(256) void finalize(const float* __restrict__ acc,
                                                const float* __restrict__ b,
                                                float* __restrict__ out,
                                                int total) {
  const int i = blockIdx.x * blockDim.x + threadIdx.x;
  if (i >= total) return;
  float v = acc[i] + b[i & (C - 1)];
  if (RELU) v = (v > 0.f) ? v : 0.f;
  out[i] = v;
}

// ---------------------------------------------------------------------------
// One full GAT layer.
// ---------------------------------------------------------------------------
template <int CIN, int COUT, bool RELU>
static void layer(const float* X, const float* W, const float* a_s,
                  const float* a_d, const float* b, const int* src,
                  const int* dst, float* H, float* ACC, float* OUT, float* ASRC,
                  float* ADST, float* M, float* DEN, float* EB, int n, int E,
                  hipStream_t stream) {
  constexpr int ROWT = 8 / (COUT / 16);
  const int nRowTiles = (n + 15) >> 4;
  gemm_wmma<CIN, COUT>
      <<<(nRowTiles + ROWT - 1) / ROWT, 256, 0, stream>>>(X, W, H, n);
  alpha_kernel<COUT>
      <<<((size_t)n * 32 + 255) / 256, 256, 0, stream>>>(H, a_s, a_d, ASRC, ADST, n);
  hipMemsetAsync(M, 0xFF, (size_t)n * sizeof(int), stream);   // "-inf" in punned order
  hipMemsetAsync(DEN, 0, (size_t)n * sizeof(float), stream);
  hipMemsetAsync(ACC, 0, (size_t)n * COUT * sizeof(float), stream);
  edge_score_max<<<(E + 255) / 256, 256, 0, stream>>>(src, dst, ASRC, ADST, EB,
                                                      (int*)M, E);
  edge_exp_sum<<<(E + 255) / 256, 256, 0, stream>>>(dst, EB, M, DEN, E);
  aggregate<COUT><<<(E + 7) / 8, 256, 0, stream>>>(src, dst, EB, DEN, H, ACC, E);
  finalize<COUT, RELU>
      <<<((size_t)n * COUT + 255) / 256, 256, 0, stream>>>(ACC, b, OUT, n * COUT);
}

extern "C" void kernel_launch(void* const* d_in, const int* in_sizes, int n_in,
                              void* d_out, int out_size, void* d_ws,
                              size_t ws_size, hipStream_t stream) {
  (void)n_in;
  (void)out_size;
  (void)ws_size;
  const float* x = (const float*)d_in[0];
  const int* ei = (const int*)d_in[1];
  const int n = in_sizes[0] / 128;  // N
  const int E = in_sizes[2];        // edge_weight length
  const int* src = ei;
  const int* dst = ei + E;
  const float* W1 = (const float*)d_in[3];
  const float* as1 = (const float*)d_in[4];
  const float* ad1 = (const float*)d_in[5];
  const float* b1 = (const float*)d_in[6];
  const float* W2 = (const float*)d_in[7];
  const float* as2 = (const float*)d_in[8];
  const float* ad2 = (const float*)d_in[9];
  const float* b2 = (const float*)d_in[10];
  const float* W3 = (const float*)d_in[11];
  const float* as3 = (const float*)d_in[12];
  const float* ad3 = (const float*)d_in[13];
  const float* b3 = (const float*)d_in[14];
  float* out = (float*)d_out;

  // Workspace carve-up (floats): H | ACC | FEAT | EB | ASRC | ADST | M | DEN
  float* ws = (float*)d_ws;
  float* H = ws;
  float* ACC = H + (size_t)n * 64;
  float* FEAT = ACC + (size_t)n * 64;
  float* EB = FEAT + (size_t)n * 64;
  float* ASRC = EB + (size_t)E;
  float* ADST = ASRC + (size_t)n;
  float* M = ADST + (size_t)n;
  float* DEN = M + (size_t)n;

  layer<128, 64, true>(x, W1, as1, ad1, b1, src, dst, H, ACC, FEAT, ASRC, ADST,
                       M, DEN, EB, n, E, stream);
  layer<64, 64, true>(FEAT, W2, as2, ad2, b2, src, dst, H, ACC, FEAT, ASRC,
                      ADST, M, DEN, EB, n, E, stream);
  layer<64, 16, false>(FEAT, W3, as3, ad3, b3, src, dst, H, ACC, out, ASRC,
                       ADST, M, DEN, EB, n, E, stream);
}